// SE_GNN_24077586661955
// MI455X (gfx1250) — compile-verified
//
#include <hip/hip_runtime.h>

typedef __attribute__((ext_vector_type(2))) float v2f;
typedef __attribute__((ext_vector_type(8))) float v8f;

#define HDIM 128

// Order-preserving float -> uint encoding so unsigned atomicMax implements float max.
__device__ __forceinline__ unsigned enc_f32(float f) {
  unsigned u = __float_as_uint(f);
  return (u & 0x80000000u) ? ~u : (u | 0x80000000u);
}
__device__ __forceinline__ float dec_f32(unsigned e) {
  return (e & 0x80000000u) ? __uint_as_float(e & 0x7FFFFFFFu)
                           : __uint_as_float(~e);
}

// ---------------------------------------------------------------------------
// Kernel 0: zero workspace (segmax encoded, segsum, neigh accumulator)
// ---------------------------------------------------------------------------
__global__ void k_init(unsigned* __restrict__ segmax, float* __restrict__ segsum,
                       float* __restrict__ neigh, int n_nodes) {
  long long i = (long long)blockIdx.x * blockDim.x + threadIdx.x;
  long long tot = (long long)n_nodes * HDIM;
  if (i < tot) neigh[i] = 0.0f;
  if (i < n_nodes) { segmax[i] = 0u; segsum[i] = 0.0f; }
}

// ---------------------------------------------------------------------------
// Kernel 1: per-edge logits + per-dst max (one wave32 per edge)
// ---------------------------------------------------------------------------
__global__ void __launch_bounds__(256)
k_logits(const float* __restrict__ ent, const float* __restrict__ rel,
         const int* __restrict__ src, const int* __restrict__ dst,
         const int* __restrict__ rid, float* __restrict__ logits,
         unsigned* __restrict__ segmax, int n_edges) {
  int wv = blockIdx.x * 8 + (threadIdx.x >> 5);   // global wave id == edge id
  int lane = threadIdx.x & 31;
  if (wv >= n_edges) return;                      // wave-uniform
  int s = src[wv], d = dst[wv], r = rid[wv];
  float4 sv = ((const float4*)(ent + (long long)s * HDIM))[lane];
  float4 rv = ((const float4*)(rel + (long long)r * HDIM))[lane];
  float4 dv = ((const float4*)(ent + (long long)d * HDIM))[lane];
  float p = (sv.x + rv.x) * dv.x + (sv.y + rv.y) * dv.y +
            (sv.z + rv.z) * dv.z + (sv.w + rv.w) * dv.w;
#pragma unroll
  for (int off = 16; off >= 1; off >>= 1) p += __shfl_xor(p, off, 32);
  if (lane == 0) {
    logits[wv] = p;
    __hip_atomic_fetch_max(&segmax[d], enc_f32(p), __ATOMIC_RELAXED,
                           __HIP_MEMORY_SCOPE_AGENT);
  }
}

// ---------------------------------------------------------------------------
// Kernel 2: ex = exp(logit - max[dst]); denom[dst] += ex   (thread per edge)
// lx is read (logit) then overwritten (ex) in place.
// ---------------------------------------------------------------------------
__global__ void k_exp(float* lx, const unsigned* __restrict__ segmax,
                      float* __restrict__ segsum, const int* __restrict__ dst,
                      int n_edges) {
  int e = blockIdx.x * blockDim.x + threadIdx.x;
  if (e >= n_edges) return;
  int d = dst[e];
  float ex = expf(lx[e] - dec_f32(segmax[d]));
  lx[e] = ex;
  __hip_atomic_fetch_add(&segsum[d], ex, __ATOMIC_RELAXED,
                         __HIP_MEMORY_SCOPE_AGENT);
}

// ---------------------------------------------------------------------------
// Kernel 3: neigh[dst] += (ent[src]+rel[rid]) * alpha  (one wave32 per edge)
// ---------------------------------------------------------------------------
__global__ void __launch_bounds__(256)
k_scatter(const float* __restrict__ ent, const float* __restrict__ rel,
          const int* __restrict__ src, const int* __restrict__ dst,
          const int* __restrict__ rid, const float* __restrict__ exbuf,
          const float* __restrict__ segsum, float* __restrict__ neigh,
          int n_edges) {
  int wv = blockIdx.x * 8 + (threadIdx.x >> 5);
  int lane = threadIdx.x & 31;
  if (wv >= n_edges) return;
  int s = src[wv], d = dst[wv], r = rid[wv];
  float alpha = exbuf[wv] / segsum[d];
  float4 sv = ((const float4*)(ent + (long long)s * HDIM))[lane];
  float4 rv = ((const float4*)(rel + (long long)r * HDIM))[lane];
  float* np = neigh + (long long)d * HDIM + lane * 4;
  __hip_atomic_fetch_add(np + 0, (sv.x + rv.x) * alpha, __ATOMIC_RELAXED, __HIP_MEMORY_SCOPE_AGENT);
  __hip_atomic_fetch_add(np + 1, (sv.y + rv.y) * alpha, __ATOMIC_RELAXED, __HIP_MEMORY_SCOPE_AGENT);
  __hip_atomic_fetch_add(np + 2, (sv.z + rv.z) * alpha, __ATOMIC_RELAXED, __HIP_MEMORY_SCOPE_AGENT);
  __hip_atomic_fetch_add(np + 3, (sv.w + rv.w) * alpha, __ATOMIC_RELAXED, __HIP_MEMORY_SCOPE_AGENT);
}

// ---------------------------------------------------------------------------
// Kernel 4: out = tanh(neigh @ W) using fp32 WMMA 16x16x4.
// One wave computes one 16-row tile across all 8 N-tiles (16x128 of output).
// W staged in LDS pair-interleaved: sB[k/2][n] = {W[k][n], W[k+1][n]} so each
// B fragment is a single aligned ds_load_b64 into an even VGPR pair.
// ---------------------------------------------------------------------------
__global__ void __launch_bounds__(256)
k_gemm_tanh(const float* __restrict__ neigh, const float* __restrict__ W,
            float* __restrict__ out, int n_nodes, int n_tiles) {
  __shared__ __align__(16) v2f sB[(HDIM / 2) * HDIM];   // 64KB
  for (int i = threadIdx.x; i < (HDIM / 2) * HDIM; i += blockDim.x) {
    int kp = i >> 7;          // k pair index: rows 2kp, 2kp+1
    int n  = i & (HDIM - 1);  // column
    v2f w;
    w.x = W[(2 * kp + 0) * HDIM + n];
    w.y = W[(2 * kp + 1) * HDIM + n];
    sB[i] = w;
  }
  __syncthreads();

  int tile = blockIdx.x * 8 + (threadIdx.x >> 5);
  if (tile >= n_tiles) return;                 // wave-uniform: EXEC stays all-1s
  int lane = threadIdx.x & 31;
  int half = lane >> 4;                        // 0 => lanes 0-15, 1 => lanes 16-31
  int l16 = lane & 15;
  int koff = half * 2;                         // A/B K sub-offset per ISA layout

  int arow = tile * 16 + l16;
  if (arow >= n_nodes) arow = n_nodes - 1;     // clamp, keep all lanes active
  const float* ap = neigh + (long long)arow * HDIM;

  v8f acc[8] = {};
  for (int k0 = 0; k0 < HDIM; k0 += 4) {
    v2f a;                                     // A 16x4 f32 layout: K=k0+koff, k0+koff+1
    a.x = ap[k0 + koff];
    a.y = ap[k0 + koff + 1];
    const v2f* bp = sB + ((k0 >> 1) + half) * HDIM;  // pair-row for this lane-half
#pragma unroll
    for (int n = 0; n < 8; ++n) {
      v2f b = bp[n * 16 + l16];                // {W[k][col], W[k+1][col]} in one b64
      acc[n] = __builtin_amdgcn_wmma_f32_16x16x4_f32(
          /*neg_a=*/false, a, /*neg_b=*/false, b,
          /*c_mod=*/(short)0, acc[n], /*reuse_a=*/false, /*reuse_b=*/false);
    }
  }

  // C/D 16x16 f32 layout: VGPR r -> row (r + half*8), col N = n*16 + l16
#pragma unroll
  for (int n = 0; n < 8; ++n) {
#pragma unroll
    for (int rr = 0; rr < 8; ++rr) {
      int m = tile * 16 + rr + half * 8;
      if (m < n_nodes)
        out[(long long)m * HDIM + n * 16 + l16] = tanhf(acc[n][rr]);
    }
  }
}

// ---------------------------------------------------------------------------
extern "C" void kernel_launch(void* const* d_in, const int* in_sizes, int n_in,
                              void* d_out, int out_size, void* d_ws, size_t ws_size,
                              hipStream_t stream) {
  const float* ent = (const float*)d_in[0];   // [N_ENT, 128]
  const float* rel = (const float*)d_in[1];   // [N_REL2, 128]
  const float* W   = (const float*)d_in[2];   // [128, 128]
  const int* src   = (const int*)d_in[3];     // [E]
  const int* dst   = (const int*)d_in[4];     // [E]
  const int* rid   = (const int*)d_in[5];     // [E]
  int n_nodes = in_sizes[0] / HDIM;
  int n_edges = in_sizes[3];

  // Workspace layout (256B-aligned slabs): segmax | segsum | ex/logits | neigh
  char* ws = (char*)d_ws;
  size_t off = 0;
  auto carve = [&](size_t bytes) {
    void* p = ws + off;
    off = (off + bytes + 255) & ~(size_t)255;
    return p;
  };
  unsigned* segmax = (unsigned*)carve((size_t)n_nodes * sizeof(unsigned));
  float* segsum    = (float*)carve((size_t)n_nodes * sizeof(float));
  float* exbuf     = (float*)carve((size_t)n_edges * sizeof(float));
  float* neigh     = (float*)carve((size_t)n_nodes * HDIM * sizeof(float));
  (void)ws_size;

  int n_tiles = (n_nodes + 15) / 16;

  long long init_elems = (long long)n_nodes * HDIM;
  k_init<<<(unsigned)((init_elems + 255) / 256), 256, 0, stream>>>(segmax, segsum, neigh, n_nodes);
  k_logits<<<(n_edges + 7) / 8, 256, 0, stream>>>(ent, rel, src, dst, rid, exbuf, segmax, n_edges);
  k_exp<<<(n_edges + 255) / 256, 256, 0, stream>>>(exbuf, segmax, segsum, dst, n_edges);
  k_scatter<<<(n_edges + 7) / 8, 256, 0, stream>>>(ent, rel, src, dst, rid, exbuf, segsum, neigh, n_edges);
  k_gemm_tanh<<<(n_tiles + 7) / 8, 256, 0, stream>>>(neigh, W, (float*)d_out, n_nodes, n_tiles);
}